// Conv2D_79216376807983
// MI455X (gfx1250) — compile-verified
//
#include <hip/hip_runtime.h>

typedef __attribute__((ext_vector_type(2))) float v2f;
typedef __attribute__((ext_vector_type(8))) float v8f;

#define IN_H 4096
#define IN_W 4096
#define KH 15
#define KW 15
#define OUT_H (IN_H - KH + 1)   // 4082
#define OUT_W (IN_W - KW + 1)   // 4082

#define HROWS     112           // output rows per wave band
#define ROWSTRIDE 17            // padded floats per LDS acc row (bank-conflict-free)
#define NWAVES    8             // waves per block (256 threads, wave32)
#define ACC_PER_WAVE    (HROWS * ROWSTRIDE + 16)  // +16 dump slots for OOB scatter
#define ROWSLOT         40                        // floats per staged row segment
#define ROWBUF_PER_WAVE (2 * ROWSLOT)             // double slot for row pairs

// One input-row step: D_r[ky][n] = sum_kx Wpad[ky][kx] * X[r][x0+n+kx]
// (16x16x16 f32 GEMM = 4 chained v_wmma_f32_16x16x4_f32), then scatter
// D_r[ky][*] into LDS accumulator row (r - ky). GUARD phases are branch-free
// via cndmask address selection so EXEC stays all-1s (WMMA requirement).
template <bool GUARD>
__device__ __forceinline__ void process_row(
    int r, int y0, int lane, int n, int half, int tb, float xv,
    float* __restrict__ rowbuf, float* __restrict__ acc, const v2f (&a)[4])
{
  rowbuf[lane] = xv;                       // ds_store_b32

  v8f c = {0.f, 0.f, 0.f, 0.f, 0.f, 0.f, 0.f, 0.f};
  #pragma unroll
  for (int j = 0; j < 4; ++j) {
    v2f bv;                                // ds_load_2addr_b32 pair
    bv.x = rowbuf[tb + 4 * j];
    bv.y = rowbuf[tb + 4 * j + 1];
    c = __builtin_amdgcn_wmma_f32_16x16x4_f32(false, a[j], false, bv,
                                              (short)0, c, false, false);
  }

  const int rb = r - y0 - 8 * half;
  if (GUARD) {
    #pragma unroll
    for (int i = 0; i < 8; ++i) {
      const int row    = rb - i;
      const int off_ok = row * ROWSTRIDE + n;
      const int m      = -(int)((unsigned)row < (unsigned)HROWS);
      const int off    = (off_ok & m) | ((HROWS * ROWSTRIDE + n) & ~m);
      atomicAdd(&acc[off], c[i]);          // ds_add_f32 (wave-private)
    }
  } else {
    const int base = (rb - 7) * ROWSTRIDE + n;
    #pragma unroll
    for (int i = 0; i < 8; ++i)
      atomicAdd(&acc[base + (7 - i) * ROWSTRIDE], c[i]);  // ds_add_f32 off:imm
  }
}

// Steady-state pair step: rows r and r+1 with two independent WMMA accumulator
// chains interleaved to cover the WMMA->WMMA C-dependency latency (§7.12.1).
__device__ __forceinline__ void process_row_pair(
    int r, int y0, int lane, int n, int half, int tb, float xv0, float xv1,
    float* __restrict__ rowbuf, float* __restrict__ acc, const v2f (&a)[4])
{
  rowbuf[lane]           = xv0;            // ds_store_b32 x2
  rowbuf[ROWSLOT + lane] = xv1;

  v8f c0 = {0.f, 0.f, 0.f, 0.f, 0.f, 0.f, 0.f, 0.f};
  v8f c1 = {0.f, 0.f, 0.f, 0.f, 0.f, 0.f, 0.f, 0.f};
  #pragma unroll
  for (int j = 0; j < 4; ++j) {
    v2f b0, b1;                            // ds_load_2addr_b32 x2
    b0.x = rowbuf[tb + 4 * j];
    b0.y = rowbuf[tb + 4 * j + 1];
    b1.x = rowbuf[ROWSLOT + tb + 4 * j];
    b1.y = rowbuf[ROWSLOT + tb + 4 * j + 1];
    c0 = __builtin_amdgcn_wmma_f32_16x16x4_f32(false, a[j], false, b0,
                                               (short)0, c0, false, false);
    c1 = __builtin_amdgcn_wmma_f32_16x16x4_f32(false, a[j], false, b1,
                                               (short)0, c1, false, false);
  }

  const int rb    = r - y0 - 8 * half;
  const int base0 = (rb - 7) * ROWSTRIDE + n;
  #pragma unroll
  for (int i = 0; i < 8; ++i)
    atomicAdd(&acc[base0 + (7 - i) * ROWSTRIDE], c0[i]);
  #pragma unroll
  for (int i = 0; i < 8; ++i)
    atomicAdd(&acc[base0 + (8 - i) * ROWSTRIDE], c1[i]);   // row r+1: +1 stride
}

__global__ __launch_bounds__(256) void conv2d_wmma_f32_kernel(
    const float* __restrict__ X, const float* __restrict__ Wgt,
    const float* __restrict__ bias, float* __restrict__ out)
{
  __shared__ float lds_acc[NWAVES * ACC_PER_WAVE];
  __shared__ float lds_row[NWAVES * ROWBUF_PER_WAVE];

  const int lane = threadIdx.x & 31;
  const int wave = threadIdx.x >> 5;
  const int n    = lane & 15;   // A-row (=ky), B/D column
  const int half = lane >> 4;   // selects K pair / M+8 per f32 WMMA layouts

  const int x0   = blockIdx.x * 16;
  const int band = blockIdx.y * NWAVES + wave;
  const int y0   = band * HROWS;

  float* acc    = lds_acc + wave * ACC_PER_WAVE;
  float* rowbuf = lds_row + wave * ROWBUF_PER_WAVE;
  for (int i = lane; i < ACC_PER_WAVE; i += 32) acc[i] = 0.0f;

  // A operand: zero-padded 16x16 weight, 4 K-chunks of 4.
  // f32 A 16x4 layout: lanes 0-15 K=0,1 in v0,v1; lanes 16-31 K=2,3; M=lane%16.
  v2f a[4];
  #pragma unroll
  for (int j = 0; j < 4; ++j) {
    const int kx0 = 4 * j + 2 * half;
    a[j].x = (n < KH && kx0     < KW) ? Wgt[n * KW + kx0    ] : 0.0f;
    a[j].y = (n < KH && kx0 + 1 < KW) ? Wgt[n * KW + kx0 + 1] : 0.0f;
  }

  const int tb    = n + 2 * half;          // B base index into rowbuf
  const int gx    = x0 + lane;
  const int xmask = (gx < IN_W) ? -1 : 0;
  const float* xcol = X + min(gx, IN_W - 1);  // per-lane clamped column base

  const int r_last = min(y0 + HROWS - 1, OUT_H - 1) + (KH - 1);
  const int pro_end    = min(y0 + KH, r_last + 1);      // first 15 rows: guarded
  const int steady_end = min(y0 + HROWS, r_last + 1);   // all targets in-band

  // Software pipeline: segments for current row(s) live in registers; next
  // row(s) load at the top of each iteration so global latency hides behind
  // the DS + WMMA work. Row index clamped (scalar s_min; r is uniform).
  #define LOAD_ROW(rr)                                                         \
    __int_as_float(__float_as_int(xcol[(size_t)min((rr), IN_H - 1) * IN_W])    \
                   & xmask)

  int r = y0;
  float xv = LOAD_ROW(r);
  for (; r < pro_end; ++r) {               // guarded prologue
    const size_t nofs = (size_t)min(r + 1, IN_H - 1) * IN_W;
    const float xn = __int_as_float(__float_as_int(xcol[nofs]) & xmask);
    __builtin_prefetch((const void*)(xcol + nofs + (size_t)15 * IN_W), 0, 1);
    process_row<true>(r, y0, lane, n, half, tb, xv, rowbuf, acc, a);
    xv = xn;
  }
  float xv1 = LOAD_ROW(r + 1);
  for (; r + 1 < steady_end; r += 2) {     // steady: row pairs, dual chains
    const size_t nofs = (size_t)min(r + 2, IN_H - 1) * IN_W;
    const float xn0 = __int_as_float(__float_as_int(xcol[nofs]) & xmask);
    const float xn1 = LOAD_ROW(r + 3);
    __builtin_prefetch((const void*)(xcol + nofs + (size_t)15 * IN_W), 0, 1);
    process_row_pair(r, y0, lane, n, half, tb, xv, xv1, rowbuf, acc, a);
    xv = xn0;
    xv1 = xn1;
  }
  for (; r < steady_end; ++r) {            // odd remainder, unguarded
    const float xn = LOAD_ROW(r + 1);
    process_row<false>(r, y0, lane, n, half, tb, xv, rowbuf, acc, a);
    xv = xn;
  }
  for (; r <= r_last; ++r) {               // guarded epilogue
    const float xn = LOAD_ROW(r + 1);
    process_row<true>(r, y0, lane, n, half, tb, xv, rowbuf, acc, a);
    xv = xn;
  }
  #undef LOAD_ROW

  __syncthreads();

  const float b0 = bias[0];
  #pragma unroll 4
  for (int rr = 0; rr < HROWS; rr += 2) {
    const int row = rr + half;
    const int y = y0 + row;
    const int x = x0 + n;
    if (y < OUT_H && x < OUT_W)
      out[(size_t)y * OUT_W + x] = acc[row * ROWSTRIDE + n] + b0;
  }
}

extern "C" void kernel_launch(void* const* d_in, const int* in_sizes, int n_in,
                              void* d_out, int out_size, void* d_ws, size_t ws_size,
                              hipStream_t stream) {
  const float* X    = (const float*)d_in[0];
  const float* Wgt  = (const float*)d_in[1];
  const float* bias = (const float*)d_in[2];
  float* out = (float*)d_out;

  const int bands = (OUT_H + HROWS - 1) / HROWS;                   // 37
  dim3 grid((OUT_W + 15) / 16, (bands + NWAVES - 1) / NWAVES, 1);  // 256 x 5
  dim3 block(256, 1, 1);
  conv2d_wmma_f32_kernel<<<grid, block, 0, stream>>>(X, Wgt, bias, out);
}